// Hamburger_23776938951025
// MI455X (gfx1250) — compile-verified
//
#include <hip/hip_runtime.h>

// ---------------------------------------------------------------------------
// Hamburger (HamNet) head for MI455X / gfx1250, fp32 end-to-end.
//
// ~290 GFLOP fp32 vs ~1.5 GB HBM traffic (~65us @ 23.3 TB/s) => compute
// bound; all contractions run on the native fp32 matrix pipe
// V_WMMA_F32_16X16X4_F32. Register blocking (4x2 / 2x4 16x16 tiles per
// wave) amortizes operand loads over 8 WMMAs per wait. The hot shared
// operand (bases, 128KB) is staged into LDS via the Tensor Data Mover
// (TENSOR_LOAD_TO_LDS + s_wait_tensorcnt), B-fragments then come from LDS.
// K=16384 contractions use split-K with a fixed-order reduction (no fp
// atomics -> deterministic).
// ---------------------------------------------------------------------------

typedef float v2f __attribute__((ext_vector_type(2)));
typedef float v8f __attribute__((ext_vector_type(8)));
typedef unsigned int u32x4 __attribute__((ext_vector_type(4)));
typedef int i32x4 __attribute__((ext_vector_type(4)));
typedef int i32x8 __attribute__((ext_vector_type(8)));

#define N_SP 16384   // H*W
#define D_CH 512     // channels (= C/S)
#define R_RK 64      // NMF rank
#define NB   8       // batch
#define EPSC 1e-6f
#define KSPLIT 16
#define KCHUNK (N_SP / KSPLIT)   // 1024
#define BCHUNK 128               // K-chunk of bases staged in LDS (32 KB)

// D = A(16x4) * B(4x16) + C, exact fp32
__device__ __forceinline__ v8f wmma4(v2f a, v2f b, v8f c) {
  return __builtin_amdgcn_wmma_f32_16x16x4_f32(false, a, false, b, (short)0, c,
                                               false, false);
}

// A fragment from row-major src[row*ld + k] (one contiguous v2f per lane).
__device__ __forceinline__ v2f ld_vec(const float* __restrict__ src, int ld,
                                      int row0, int k, int col, int half) {
  return *(const v2f*)(src + (size_t)(row0 + col) * ld + k + 2 * half);
}

// A-transposed / B-row-major fragment: element [k',c] = src[k'*ld + c0+col].
__device__ __forceinline__ v2f ld_str(const float* __restrict__ src, int ld,
                                      int c0, int k, int col, int half) {
  const float* p = src + (size_t)(k + 2 * half) * ld + c0 + col;
  v2f r;
  r.x = p[0];
  r.y = p[ld];
  return r;
}

// Same fragment shape, from LDS.
__device__ __forceinline__ v2f ld_lds(const float* src, int ld, int c0, int k,
                                      int col, int half) {
  const float* p = src + (k + 2 * half) * ld + c0 + col;
  v2f r;
  r.x = p[0];
  r.y = p[ld];
  return r;
}

// ---------------------------------------------------------------------------
// TDM: DMA a rows x cols fp32 tile (row stride = srcStride elements) from
// global memory into LDS at byte offset ldsOff.  D# packed per ISA 8.3/8.4:
// group0 = {count=1 | lds_addr | global_addr[56:0] | type=2},
// group1 = {data_size=4B, tensor_dim0/1 = tile_dim0/1 = cols/rows, stride0}.
// Groups 2/3 and the trailing descriptor words are unused for 2D tiles.
// This toolchain exposes the 6-arg builtin form.
// ---------------------------------------------------------------------------
__device__ __forceinline__ void tdm_load_2d(const float* gsrc, unsigned ldsOff,
                                            unsigned rows, unsigned cols,
                                            unsigned srcStride) {
  unsigned long long ga = (unsigned long long)(uintptr_t)gsrc;
  u32x4 g0;
  g0[0] = 1u;                                               // count=1 (valid)
  g0[1] = ldsOff;                                           // lds_addr (bytes)
  g0[2] = (unsigned)ga;                                     // global_addr lo
  g0[3] = (unsigned)((ga >> 32) & 0x01FFFFFFu) | (2u << 30);  // hi + type=2
  i32x8 g1;
  g1[0] = (int)(2u << 16);                                  // data_size = 4B
  g1[1] = (int)((cols & 0xFFFFu) << 16);                    // tensor_dim0 lo
  g1[2] = (int)(((cols >> 16) & 0xFFFFu) | ((rows & 0xFFFFu) << 16));
  g1[3] = (int)(((rows >> 16) & 0xFFFFu) | ((cols & 0xFFFFu) << 16)); // tile0
  g1[4] = (int)(rows & 0xFFFFu);                            // tile_dim1
  g1[5] = (int)srcStride;                                   // dim0_stride lo
  g1[6] = 0;
  g1[7] = 0;
  i32x4 z4 = {0, 0, 0, 0};
  i32x8 z8 = {0, 0, 0, 0, 0, 0, 0, 0};
  __builtin_amdgcn_tensor_load_to_lds(g0, g1, z4, z4, z8, 0);
}

// ---------------------------------------------------------------------------
// conv1x1 (+bias, optional residual, ReLU): out[b,o,n] = relu(resid +
//   sum_c W[o,c]*X[b,c,n] + bias[o]).
// Wave computes 64(M)x32(N): 4x2 tiles, 8 WMMA per k-step behind one wait.
// grid(N/128, O/64, B), block 128.
// ---------------------------------------------------------------------------
__global__ void k_conv1x1(const float* __restrict__ W,
                          const float* __restrict__ X,
                          const float* __restrict__ bias,
                          const float* __restrict__ resid,
                          float* __restrict__ out) {
  const int lane = threadIdx.x & 31, wv = threadIdx.x >> 5;
  const int col = lane & 15, half = lane >> 4;
  const int b  = blockIdx.z;
  const int m0 = blockIdx.y * 64;
  const int n0 = blockIdx.x * 128 + wv * 32;
  const float* Xb = X + (size_t)b * D_CH * N_SP;

  v8f acc[4][2] = {};
  for (int k = 0; k < D_CH; k += 4) {
    if (k + 64 < D_CH)  // stream-ahead hint -> global_prefetch_b8
      __builtin_prefetch(Xb + (size_t)(k + 64) * N_SP + n0 + col, 0, 1);
    v2f a[4], bb[2];
#pragma unroll
    for (int i = 0; i < 4; ++i)
      a[i] = ld_vec(W, D_CH, m0 + 16 * i, k, col, half);  // W row-major [o,c]
#pragma unroll
    for (int j = 0; j < 2; ++j)
      bb[j] = ld_str(Xb, N_SP, n0 + 16 * j, k, col, half);  // X [c,n]
#pragma unroll
    for (int i = 0; i < 4; ++i)
#pragma unroll
      for (int j = 0; j < 2; ++j) acc[i][j] = wmma4(a[i], bb[j], acc[i][j]);
  }
#pragma unroll
  for (int i = 0; i < 4; ++i)
#pragma unroll
    for (int j = 0; j < 2; ++j)
#pragma unroll
      for (int v = 0; v < 8; ++v) {
        int row = m0 + 16 * i + v + 8 * half;
        size_t idx = ((size_t)b * D_CH + row) * N_SP + n0 + 16 * j + col;
        float val = acc[i][j][v] + bias[row];
        if (resid) val += resid[idx];
        out[idx] = fmaxf(val, 0.f);
      }
}

// Broadcast shared bases [D,R] to per-batch working copies.
__global__ void k_bcast_bases(const float* __restrict__ src,
                              float* __restrict__ dst) {
  int i = blockIdx.x * blockDim.x + threadIdx.x;
  const int tot = D_CH * R_RK;
  if (i < tot) {
    float v = src[i];
#pragma unroll
    for (int b = 0; b < NB; ++b) dst[(size_t)b * tot + i] = v;
  }
}

// ---------------------------------------------------------------------------
// num[b,n,r] = sum_d h[b,d,n] * bases[b,d,r].
// bases K-chunks (128x64 = 32KB) staged to LDS by wave 0 via the Tensor Data
// Mover; all 4 waves read B-fragments from LDS. Wave computes 32(n)x64(r).
// grid(N/128, 1, B), block 128.
// ---------------------------------------------------------------------------
__global__ void k_coef_num(const float* __restrict__ h,
                           const float* __restrict__ basesW,
                           float* __restrict__ num) {
  __shared__ float sB[BCHUNK * R_RK];  // 32 KB
  const int lane = threadIdx.x & 31, wv = threadIdx.x >> 5;
  const int col = lane & 15, half = lane >> 4;
  const int b  = blockIdx.z;
  const int m0 = blockIdx.x * 128 + wv * 32;  // n rows
  const float* hb = h + (size_t)b * D_CH * N_SP;
  const float* Bb = basesW + (size_t)b * D_CH * R_RK;
  const unsigned sOff = (unsigned)(uintptr_t)(void*)sB;

  v8f acc[2][4] = {};
  for (int kc = 0; kc < D_CH; kc += BCHUNK) {
    __syncthreads();  // previous chunk fully consumed
    if (wv == 0) {
      tdm_load_2d(Bb + (size_t)kc * R_RK, sOff, BCHUNK, R_RK, R_RK);
      __builtin_amdgcn_s_wait_tensorcnt(0);
    }
    __syncthreads();  // chunk visible to all waves
    for (int k = 0; k < BCHUNK; k += 4) {
      v2f a0 = ld_str(hb, N_SP, m0, kc + k, col, half);       // h[d,n]^T
      v2f a1 = ld_str(hb, N_SP, m0 + 16, kc + k, col, half);
#pragma unroll
      for (int s = 0; s < 4; ++s) {
        v2f bb = ld_lds(sB, R_RK, s * 16, k, col, half);
        acc[0][s] = wmma4(a0, bb, acc[0][s]);
        acc[1][s] = wmma4(a1, bb, acc[1][s]);
      }
    }
  }
  float* nb = num + (size_t)b * N_SP * R_RK;
#pragma unroll
  for (int i = 0; i < 2; ++i)
#pragma unroll
    for (int s = 0; s < 4; ++s)
#pragma unroll
      for (int v = 0; v < 8; ++v)
        nb[(size_t)(m0 + 16 * i + v + 8 * half) * R_RK + s * 16 + col] =
            acc[i][s][v];
}

// coef[row,:] = softmax(num[row,:64]); one wave32 per row.
__global__ void k_softmax(const float* __restrict__ num,
                          float* __restrict__ coef) {
  const int lane = threadIdx.x & 31;
  const size_t row = (size_t)blockIdx.x * 8 + (threadIdx.x >> 5);
  const float* p = num + row * R_RK;
  float v0 = p[lane], v1 = p[lane + 32];
  float m = fmaxf(v0, v1);
#pragma unroll
  for (int off = 16; off > 0; off >>= 1) m = fmaxf(m, __shfl_xor(m, off, 32));
  float e0 = expf(v0 - m), e1 = expf(v1 - m);
  float s = e0 + e1;
#pragma unroll
  for (int off = 16; off > 0; off >>= 1) s += __shfl_xor(s, off, 32);
  float r = 1.f / s;
  float* q = coef + row * R_RK;
  q[lane] = e0 * r;
  q[lane + 32] = e1 * r;
}

// btb[b,r,s] = sum_d bases[d,r]*bases[d,s].  grid(B), block 128.
__global__ void k_btb(const float* __restrict__ basesW,
                      float* __restrict__ btb) {
  const int lane = threadIdx.x & 31, wv = threadIdx.x >> 5;
  const int col = lane & 15, half = lane >> 4;
  const int b = blockIdx.x, m0 = wv * 16;
  const float* Bb = basesW + (size_t)b * D_CH * R_RK;

  v8f acc[4] = {};
  for (int k = 0; k < D_CH; k += 4) {
    v2f a = ld_str(Bb, R_RK, m0, k, col, half);  // A[r,d] = bases[d,r]
#pragma unroll
    for (int s = 0; s < 4; ++s) {
      v2f bb = ld_str(Bb, R_RK, s * 16, k, col, half);
      acc[s] = wmma4(a, bb, acc[s]);
    }
  }
  float* ob = btb + (size_t)b * R_RK * R_RK;
#pragma unroll
  for (int s = 0; s < 4; ++s)
#pragma unroll
    for (int v = 0; v < 8; ++v)
      ob[(m0 + v + 8 * half) * R_RK + s * 16 + col] = acc[s][v];
}

// den = coef @ btb (K=64); coef = coef*num/(den+eps).  grid(N/64,1,B).
__global__ void k_update_coef(float* __restrict__ coef,
                              const float* __restrict__ num,
                              const float* __restrict__ btb) {
  const int lane = threadIdx.x & 31, wv = threadIdx.x >> 5;
  const int col = lane & 15, half = lane >> 4;
  const int b  = blockIdx.z;
  const int m0 = blockIdx.x * 64 + wv * 16;
  float* cb = coef + (size_t)b * N_SP * R_RK;
  const float* nb = num + (size_t)b * N_SP * R_RK;
  const float* tb = btb + (size_t)b * R_RK * R_RK;

  v8f acc[4] = {};
  for (int k = 0; k < R_RK; k += 4) {
    v2f a = ld_vec(cb, R_RK, m0, k, col, half);  // coef row-major [n,r]
#pragma unroll
    for (int s = 0; s < 4; ++s) {
      v2f bb = ld_str(tb, R_RK, s * 16, k, col, half);
      acc[s] = wmma4(a, bb, acc[s]);
    }
  }
#pragma unroll
  for (int s = 0; s < 4; ++s)
#pragma unroll
    for (int v = 0; v < 8; ++v) {
      size_t idx = (size_t)(m0 + v + 8 * half) * R_RK + s * 16 + col;
      cb[idx] = cb[idx] * nb[idx] / (acc[s][v] + EPSC);
    }
}

// Split-K partials of num2[b,d,r] = sum_n h[d,n]*coef[n,r].
// Wave computes 32(d)x64(r): 2x4 tiles.  grid(KSPLIT, D/128, B), block 128.
__global__ void k_num2_split(const float* __restrict__ h,
                             const float* __restrict__ coef,
                             float* __restrict__ part) {
  const int lane = threadIdx.x & 31, wv = threadIdx.x >> 5;
  const int col = lane & 15, half = lane >> 4;
  const int b  = blockIdx.z;
  const int m0 = blockIdx.y * 128 + wv * 32;  // d rows
  const int kz = blockIdx.x, kb = kz * KCHUNK;
  const float* hb = h + (size_t)b * D_CH * N_SP;
  const float* cb = coef + (size_t)b * N_SP * R_RK;

  v8f acc[2][4] = {};
  for (int k = kb; k < kb + KCHUNK; k += 4) {
    v2f a0 = ld_vec(hb, N_SP, m0, k, col, half);  // h row-major [d,n]
    v2f a1 = ld_vec(hb, N_SP, m0 + 16, k, col, half);
#pragma unroll
    for (int s = 0; s < 4; ++s) {
      v2f bb = ld_str(cb, R_RK, s * 16, k, col, half);  // B[n,r]
      acc[0][s] = wmma4(a0, bb, acc[0][s]);
      acc[1][s] = wmma4(a1, bb, acc[1][s]);
    }
  }
  float* pb = part + ((size_t)b * KSPLIT + kz) * D_CH * R_RK;
#pragma unroll
  for (int i = 0; i < 2; ++i)
#pragma unroll
    for (int s = 0; s < 4; ++s)
#pragma unroll
      for (int v = 0; v < 8; ++v)
        pb[(size_t)(m0 + 16 * i + v + 8 * half) * R_RK + s * 16 + col] =
            acc[i][s][v];
}

// Split-K partials of ctc[b,r,s] = sum_n coef[n,r]*coef[n,s].
// grid(KSPLIT, 1, B), block 128.
__global__ void k_ctc_split(const float* __restrict__ coef,
                            float* __restrict__ part) {
  const int lane = threadIdx.x & 31, wv = threadIdx.x >> 5;
  const int col = lane & 15, half = lane >> 4;
  const int b  = blockIdx.z;
  const int m0 = wv * 16;
  const int kz = blockIdx.x, kb = kz * KCHUNK;
  const float* cb = coef + (size_t)b * N_SP * R_RK;

  v8f acc[4] = {};
  for (int k = kb; k < kb + KCHUNK; k += 4) {
    v2f a = ld_str(cb, R_RK, m0, k, col, half);  // A[r,n] = coef[n,r]
#pragma unroll
    for (int s = 0; s < 4; ++s) {
      v2f bb = ld_str(cb, R_RK, s * 16, k, col, half);
      acc[s] = wmma4(a, bb, acc[s]);
    }
  }
  float* pb = part + ((size_t)b * KSPLIT + kz) * R_RK * R_RK;
#pragma unroll
  for (int s = 0; s < 4; ++s)
#pragma unroll
    for (int v = 0; v < 8; ++v)
      pb[(m0 + v + 8 * half) * R_RK + s * 16 + col] = acc[s][v];
}

// Deterministic fixed-order split-K reduction: out[b,i]=sum_kz part[b,kz,i].
__global__ void k_reduce(const float* __restrict__ part,
                         float* __restrict__ out, int len) {
  size_t i = (size_t)blockIdx.x * blockDim.x + threadIdx.x;
  size_t b = i / len, j = i % len;
  float s = 0.f;
  for (int kz = 0; kz < KSPLIT; ++kz)
    s += part[((size_t)b * KSPLIT + kz) * len + j];
  out[(size_t)b * len + j] = s;
}

// den2 = bases @ ctc; bases = bases*num2/(den2+eps).  grid(D/64,1,B).
__global__ void k_bases_update(float* __restrict__ basesW,
                               const float* __restrict__ ctc,
                               const float* __restrict__ num2) {
  const int lane = threadIdx.x & 31, wv = threadIdx.x >> 5;
  const int col = lane & 15, half = lane >> 4;
  const int b  = blockIdx.z;
  const int m0 = blockIdx.x * 64 + wv * 16;  // d rows
  float* Bb = basesW + (size_t)b * D_CH * R_RK;
  const float* tb = ctc + (size_t)b * R_RK * R_RK;
  const float* nb = num2 + (size_t)b * D_CH * R_RK;

  v8f acc[4] = {};
  for (int k = 0; k < R_RK; k += 4) {
    v2f a = ld_vec(Bb, R_RK, m0, k, col, half);  // bases row-major [d,r]
#pragma unroll
    for (int s = 0; s < 4; ++s) {
      v2f bb = ld_str(tb, R_RK, s * 16, k, col, half);
      acc[s] = wmma4(a, bb, acc[s]);
    }
  }
#pragma unroll
  for (int s = 0; s < 4; ++s)
#pragma unroll
    for (int v = 0; v < 8; ++v) {
      size_t idx = (size_t)(m0 + v + 8 * half) * R_RK + s * 16 + col;
      Bb[idx] = Bb[idx] * nb[idx] / (acc[s][v] + EPSC);
    }
}

// y[b,d,n] = sum_r bases[d,r]*coef[n,r]  (bases @ coef^T).  K=64.
// grid(N/256, D/16, B), block 128: wave covers 16x64 output.
__global__ void k_recon(const float* __restrict__ basesW,
                        const float* __restrict__ coef,
                        float* __restrict__ y) {
  const int lane = threadIdx.x & 31, wv = threadIdx.x >> 5;
  const int col = lane & 15, half = lane >> 4;
  const int b  = blockIdx.z;
  const int m0 = blockIdx.y * 16;              // d
  const int n0 = blockIdx.x * 256 + wv * 64;   // n macro per wave
  const float* Bb = basesW + (size_t)b * D_CH * R_RK;
  const float* cb = coef + (size_t)b * N_SP * R_RK;

  v8f acc[4] = {};
  for (int k = 0; k < R_RK; k += 4) {
    v2f a = ld_vec(Bb, R_RK, m0, k, col, half);
#pragma unroll
    for (int t = 0; t < 4; ++t) {
      // B[r,n] = coef[n,r]: contiguous v2f along r
      v2f bb = ld_vec(cb, R_RK, n0 + t * 16, k, col, half);
      acc[t] = wmma4(a, bb, acc[t]);
    }
  }
  float* yb = y + (size_t)b * D_CH * N_SP;
#pragma unroll
  for (int t = 0; t < 4; ++t)
#pragma unroll
    for (int v = 0; v < 8; ++v)
      yb[(size_t)(m0 + v + 8 * half) * N_SP + n0 + t * 16 + col] = acc[t][v];
}

// ---------------------------------------------------------------------------
extern "C" void kernel_launch(void* const* d_in, const int* in_sizes, int n_in,
                              void* d_out, int out_size, void* d_ws,
                              size_t ws_size, hipStream_t stream) {
  (void)in_sizes; (void)n_in; (void)out_size; (void)ws_size;
  const float* x     = (const float*)d_in[0];
  const float* w_in  = (const float*)d_in[1];
  const float* b_in  = (const float*)d_in[2];
  const float* w_out = (const float*)d_in[3];
  const float* b_out = (const float*)d_in[4];
  const float* bases = (const float*)d_in[5];
  float* out = (float*)d_out;

  float* ws = (float*)d_ws;
  size_t o = 0;
  float* h      = ws + o; o += (size_t)NB * D_CH * N_SP;   // 256 MB, reused as y
  float* coef   = ws + o; o += (size_t)NB * N_SP * R_RK;   // 32 MB
  float* num    = ws + o; o += (size_t)NB * N_SP * R_RK;   // 32 MB
  float* basesW = ws + o; o += (size_t)NB * D_CH * R_RK;   // 1 MB
  float* btb    = ws + o; o += (size_t)NB * R_RK * R_RK;
  float* ctc    = ws + o; o += (size_t)NB * R_RK * R_RK;
  float* num2   = ws + o; o += (size_t)NB * D_CH * R_RK;
  float* pnum2  = ws + o; o += (size_t)NB * KSPLIT * D_CH * R_RK;  // 16 MB
  float* pctc   = ws + o; o += (size_t)NB * KSPLIT * R_RK * R_RK;  // 2 MB

  const dim3 blk(128);
  const dim3 gConv(N_SP / 128, D_CH / 64, NB);
  const dim3 gCN(N_SP / 128, 1, NB);
  const dim3 gUC(N_SP / 64, 1, NB);
  const dim3 gBU(D_CH / 64, 1, NB);

  // ham_in: h = relu(W_in x + b_in)
  k_conv1x1<<<gConv, blk, 0, stream>>>(w_in, x, b_in, nullptr, h);
  k_bcast_bases<<<(D_CH * R_RK + 255) / 256, 256, 0, stream>>>(bases, basesW);

  // coef = softmax(x^T bases)
  k_coef_num<<<gCN, blk, 0, stream>>>(h, basesW, num);
  k_softmax<<<NB * N_SP / 8, 256, 0, stream>>>(num, coef);

  for (int it = 0; it < 7; ++it) {
    // coef = coef * (x^T b) / (coef (b^T b) + eps)
    k_coef_num<<<gCN, blk, 0, stream>>>(h, basesW, num);
    k_btb<<<dim3(NB), blk, 0, stream>>>(basesW, btb);
    k_update_coef<<<gUC, blk, 0, stream>>>(coef, num, btb);
    // bases = bases * (x coef) / (bases (coef^T coef) + eps)
    k_num2_split<<<dim3(KSPLIT, D_CH / 128, NB), blk, 0, stream>>>(h, coef,
                                                                   pnum2);
    k_reduce<<<NB * D_CH * R_RK / 256, 256, 0, stream>>>(pnum2, num2,
                                                         D_CH * R_RK);
    k_ctc_split<<<dim3(KSPLIT, 1, NB), blk, 0, stream>>>(coef, pctc);
    k_reduce<<<NB * R_RK * R_RK / 256, 256, 0, stream>>>(pctc, ctc,
                                                         R_RK * R_RK);
    k_bases_update<<<gBU, blk, 0, stream>>>(basesW, ctc, num2);
  }

  // compute_coef
  k_coef_num<<<gCN, blk, 0, stream>>>(h, basesW, num);
  k_btb<<<dim3(NB), blk, 0, stream>>>(basesW, btb);
  k_update_coef<<<gUC, blk, 0, stream>>>(coef, num, btb);

  // y = bases @ coef^T   (overwrites h, which is dead now)
  k_recon<<<dim3(N_SP / 256, D_CH / 16, NB), blk, 0, stream>>>(basesW, coef, h);

  // ham_out + residual relu
  k_conv1x1<<<gConv, blk, 0, stream>>>(w_out, h, b_out, x, out);
}